// CRFLoss_43946105372757
// MI455X (gfx1250) — compile-verified
//
#include <hip/hip_runtime.h>
#include <stdint.h>

// CRF loss: B=64, S=512, T=48, START=46, END=47, eps=1e-12
#define BSZ 64
#define SEQ 512
#define NT 48
#define START_TAG 46
#define END_TAG 47
#define TPB 384                 // 8 row-groups x 48 columns = 12 wave32s
#define KG 8                    // row groups
#define ROWS_PER 6              // 48 / 8 rows per thread
#define DEPTH 16                // async pipeline depth (tiles in flight)
#define TILE_ELEMS (NT * NT)    // 2304 floats
#define CHUNK_BYTES 8           // one b64 async chunk per lane
#define CHUNKS 3                // 384 thr * 3 * 8B = 9216B = one tile
// per-wave async instrs per tile = CHUNKS; steady-state outstanding = 3*DEPTH=48
#define WAIT_STEADY "s_wait_asynccnt 45"   // 3*(DEPTH-1)

// Issue an async global->LDS copy of one 9216B transition tile.
// 3 x global_load_async_to_lds_b64 per thread; per wave-instruction the 32
// lanes cover a contiguous 256B span (fully coalesced). Tracked by ASYNCcnt.
__device__ __forceinline__ void async_tile_load(const float* __restrict__ gsrc,
                                                const float* ldst) {
    // Generic pointers to __shared__ are {shared_aperture_hi, lds_offset}:
    // low 32 bits are the workgroup-relative LDS byte offset that the async
    // instruction expects in its LDS-address VGPR.
    uint64_t gaddr = (uint64_t)(const void*)gsrc;
    unsigned laddr = (unsigned)(uint64_t)(const void*)ldst;
    unsigned t8 = (unsigned)threadIdx.x * (unsigned)CHUNK_BYTES;
#pragma unroll
    for (int r = 0; r < CHUNKS; ++r) {
        unsigned off = t8 + (unsigned)(r * (TPB * CHUNK_BYTES));
        unsigned l = laddr + off;
        uint64_t g = gaddr + (uint64_t)off;
        asm volatile("global_load_async_to_lds_b64 %0, %1, off"
                     :: "v"(l), "v"(g)
                     : "memory");
    }
}

__global__ __launch_bounds__(TPB) void crf_forward_kernel(
    const float* __restrict__ trans,   // (64, 512, 48, 48) f32
    const int*   __restrict__ target,  // (64, 512) int
    float* __restrict__ ws)            // per-batch losses (64)
{
    __shared__ float tiles[DEPTH][TILE_ELEMS];  // 147456 B ring of step tiles
    __shared__ float s_forw[NT];
    __shared__ float s_pm[KG][NT];
    __shared__ float s_ps[KG][NT];
    __shared__ float s_red[512];

    const int tid = threadIdx.x;
    const int b   = blockIdx.x;
    const int k   = tid / NT;        // row group 0..7
    const int j   = tid - k * NT;    // column 0..47
    const float* tb = trans + (size_t)b * (size_t)SEQ * TILE_ELEMS;

    // ---- prologue: fill the whole DEPTH-deep async pipe (tiles 1..DEPTH) ----
#pragma unroll
    for (int t = 1; t <= DEPTH; ++t)
        async_tile_load(tb + (size_t)t * TILE_ELEMS, &tiles[(t - 1) % DEPTH][0]);

    // forw0 = transition[b, 0, START, :]
    if (tid < NT) s_forw[tid] = tb[START_TAG * NT + tid];

    // ---- target score (overlaps the in-flight DMA) ----
    float ts = 0.f;
    for (int s = tid; s < SEQ; s += TPB) {
        int tgt  = target[b * SEQ + s];
        int prev = (s == 0) ? START_TAG : target[b * SEQ + s - 1];
        ts += tb[(size_t)s * TILE_ELEMS + prev * NT + tgt];
    }
    s_red[tid] = ts;
    if (tid < 512 - TPB) s_red[tid + TPB] = 0.f;
    __syncthreads();
#pragma unroll
    for (int off = 256; off > 0; off >>= 1) {
        if (tid < off) s_red[tid] += s_red[tid + off];
        __syncthreads();
    }
    // s_red[0] == target_score(b); stays untouched below.

    // establish loop invariant: tile 1 resident, forw0 visible
    asm volatile(WAIT_STEADY ::: "memory");
    __syncthreads();

    // ---- sequential logsumexp recursion over s = 1..511 (2 barriers/step) ----
    for (int s = 1; s < SEQ; ++s) {
        const int slot = (s - 1) & (DEPTH - 1);
        const float* tile = &tiles[slot][0];

        float v[ROWS_PER];
        float m = -3.402823466e38f;
#pragma unroll
        for (int r = 0; r < ROWS_PER; ++r) {
            const int i = k * ROWS_PER + r;
            float t = s_forw[i] + tile[i * NT + j];
            v[r] = t;
            m = fmaxf(m, t);
        }
        float sum = 0.f;
#pragma unroll
        for (int r = 0; r < ROWS_PER; ++r) sum += __expf(v[r] - m);
        s_pm[k][j] = m;
        s_ps[k][j] = sum;
        __syncthreads();   // B: partials visible; all reads of tile s drained

        if (k == 0) {      // 48 threads combine the 8 partials per column
            float M = s_pm[0][j];
#pragma unroll
            for (int kk = 1; kk < KG; ++kk) M = fmaxf(M, s_pm[kk][j]);
            float S = 0.f;
#pragma unroll
            for (int kk = 0; kk < KG; ++kk)
                S += s_ps[kk][j] * __expf(s_pm[kk][j] - M);
            s_forw[j] = __logf(S + 1e-12f) + M;   // log(sum(exp(.-m))+eps)+m
        }

        // refill the slot just consumed with tile s+DEPTH (overlaps the reduce)
        if (s + DEPTH < SEQ) {
            async_tile_load(tb + (size_t)(s + DEPTH) * TILE_ELEMS,
                            &tiles[slot][0]);
            // outstanding: tiles s+1..s+DEPTH = 48 instrs; <=45 left means
            // tile s+1 is complete (async loads complete in order)
            asm volatile(WAIT_STEADY ::: "memory");
        } else {
            asm volatile("s_wait_asynccnt 0" ::: "memory");
        }
        __syncthreads();   // C: new forw AND tile s+1 visible
    }

    if (tid == 0) ws[b] = s_forw[END_TAG] - s_red[0];  // -score + forw[END]
}

__global__ __launch_bounds__(BSZ) void crf_finalize_kernel(
    const float* __restrict__ ws, float* __restrict__ out)
{
    __shared__ float r[BSZ];
    const int t = threadIdx.x;
    r[t] = ws[t];
    __syncthreads();
#pragma unroll
    for (int off = BSZ / 2; off > 0; off >>= 1) {
        if (t < off) r[t] += r[t + off];
        __syncthreads();
    }
    if (t == 0) out[0] = r[0] * (1.0f / (float)BSZ);
}

extern "C" void kernel_launch(void* const* d_in, const int* in_sizes, int n_in,
                              void* d_out, int out_size, void* d_ws, size_t ws_size,
                              hipStream_t stream) {
    (void)in_sizes; (void)n_in; (void)out_size; (void)ws_size;
    const float* trans  = (const float*)d_in[0];   // f32 (64,512,48,48)
    const int*   target = (const int*)d_in[1];     // integer input -> const int*
    float* ws = (float*)d_ws;                      // 64 per-batch losses

    crf_forward_kernel<<<BSZ, TPB, 0, stream>>>(trans, target, ws);
    crf_finalize_kernel<<<1, BSZ, 0, stream>>>(ws, (float*)d_out);
}